// RiemannianNetwork_18545668784623
// MI455X (gfx1250) — compile-verified
//
#include <hip/hip_runtime.h>

#define DIMQ 7
#define WIDTH 128
#define TB 64          // batch samples per workgroup
#define LDW 132        // padded LDS leading dim (floats); 132*4=528B, 16B-aligned rows
#define EPSV 1e-5f

typedef float v2f __attribute__((ext_vector_type(2)));
typedef float v8f __attribute__((ext_vector_type(8)));

// ---- CDNA5 WMMA f32 16x16x4 fragment helpers (wave32 layouts, ISA 7.12.2) ----
// A (16x4 MxK):  lane l, vgpr j -> M = l&15, K = 2*(l>>4)+j
// B (4x16 KxN):  lane l, vgpr j -> N = l&15, K = 2*(l>>4)+j
// C/D (16x16):   lane l, vgpr v -> N = l&15, M = v + 8*(l>>4)
__device__ __forceinline__ v2f fragA(const float* X, int ld, int m0, int k0, int lane) {
  const int m = m0 + (lane & 15);
  const int k = k0 + ((lane >> 4) << 1);
  v2f a; a.x = X[m * ld + k]; a.y = X[m * ld + k + 1]; return a;
}
// B[k][n] = W[n][k]  (W row-major [N][K]) -> computes X @ W^T
__device__ __forceinline__ v2f fragBT(const float* W, int ld, int n0, int k0, int lane) {
  const int n = n0 + (lane & 15);
  const int k = k0 + ((lane >> 4) << 1);
  v2f b; b.x = W[n * ld + k]; b.y = W[n * ld + k + 1]; return b;
}
// B[k][n] = W[k][n]  (W row-major [K][N]) -> computes X @ W
__device__ __forceinline__ v2f fragBN(const float* W, int ld, int k0, int n0, int lane) {
  const int n = n0 + (lane & 15);
  const int k = k0 + ((lane >> 4) << 1);
  v2f b; b.x = W[k * ld + n]; b.y = W[(k + 1) * ld + n]; return b;
}
#define WMMA_F32(a, b, c) \
  __builtin_amdgcn_wmma_f32_16x16x4_f32(false, (a), false, (b), (short)0, (c), false, false)

__global__ __launch_bounds__(256) void delan_fused(
    const float* __restrict__ q, const float* __restrict__ qd,
    const float* __restrict__ W0, const float* __restrict__ b0,
    const float* __restrict__ W1, const float* __restrict__ b1,
    const float* __restrict__ Wlo, const float* __restrict__ blo,
    const float* __restrict__ Wld, const float* __restrict__ bld,
    float* __restrict__ outM, float* __restrict__ outMi,
    float* __restrict__ outC, float* __restrict__ outT, int Btot)
{
  extern __shared__ float smem[];
  float* sW1 = smem;                 // 128 x LDW   (W1[o][i])
  float* sW0 = sW1 + 128 * LDW;      // 128 x 16    (W0[o][k], k>=7 zero)
  float* sWc = sW0 + 128 * 16;       // 32  x LDW   (rows 0-6 Wld, 7-27 Wlo, 28-31 zero)
  float* sIn = sWc + 32 * LDW;       // 128 x 8     (rows 0-63 q, 64-127 qd, col7 zero)
  float* sA  = sIn + 128 * 8;        // 128 x LDW   (y0|h0, later rows 64+: t)
  float* sB  = sA + 128 * LDW;       // 128 x LDW   (y1|t1, later rows 64+: z)
  float* sH  = sB + 128 * LDW;       // 128 x 32    (head fwd | ldt, later rows 64+: w')
  float* sMd = sH + 128 * 32;        // 64 x 8      (Mdt_qd)
  float* sb0 = sMd + 64 * 8;         // 128
  float* sb1 = sb0 + 128;            // 128
  float* sbc = sb1 + 128;            // 32

  const int tid  = threadIdx.x;
  const int lane = tid & 31;
  const int w    = tid >> 5;
  const int gb   = blockIdx.x * TB;
  const v8f z8   = {0.f,0.f,0.f,0.f,0.f,0.f,0.f,0.f};

  // ================= load weights / biases / inputs into LDS =================
  {
    const float4* W1v = (const float4*)W1;            // 128x128, rows 16B-aligned
    for (int idx = tid; idx < 128 * 32; idx += 256) {
      int o = idx >> 5, i4 = idx & 31;
      *(float4*)(sW1 + o * LDW + 4 * i4) = W1v[idx];
    }
  }
  for (int idx = tid; idx < 128 * 16; idx += 256) {
    int o = idx >> 4, k = idx & 15;
    sW0[idx] = (k < DIMQ) ? W0[o * DIMQ + k] : 0.f;
  }
  for (int idx = tid; idx < 32 * 128; idx += 256) {
    int r = idx >> 7, c = idx & 127;
    float v = 0.f;
    if (r < 7) v = Wld[r * 128 + c];
    else if (r < 28) v = Wlo[(r - 7) * 128 + c];
    sWc[r * LDW + c] = v;
  }
  if (tid < 128) { sb0[tid] = b0[tid]; sb1[tid] = b1[tid]; }
  if (tid < 32)  sbc[tid] = (tid < 7) ? bld[tid] : ((tid < 28) ? blo[tid - 7] : 0.f);
  for (int idx = tid; idx < TB * 8; idx += 256) {
    int s = idx >> 3, k = idx & 7;
    bool ok = (k < DIMQ) && (gb + s < Btot);
    sIn[idx]          = ok ? q [(size_t)(gb + s) * DIMQ + k] : 0.f;
    sIn[TB * 8 + idx] = ok ? qd[(size_t)(gb + s) * DIMQ + k] : 0.f;
  }
  __syncthreads();

  // ============ Phase A: [q ; qd] (128x8) @ W0^T -> a0 | v0 (128x128) ========
  {
    v8f acc[8]; for (int nt = 0; nt < 8; ++nt) acc[nt] = z8;
    #pragma unroll
    for (int ks = 0; ks < 2; ++ks) {
      v2f a = fragA(sIn, 8, w * 16, ks * 4, lane);
      #pragma unroll
      for (int nt = 0; nt < 8; ++nt) {
        v2f b = fragBT(sW0, 16, nt * 16, ks * 4, lane);
        acc[nt] = WMMA_F32(a, b, acc[nt]);
      }
    }
    const int mb = w * 16 + 8 * (lane >> 4), nb = lane & 15;
    if (w < 4) {                                      // forward rows: y0 = relu(a0+b0)
      #pragma unroll
      for (int nt = 0; nt < 8; ++nt) {
        const int n = nt * 16 + nb;
        const float bias = sb0[n];
        #pragma unroll
        for (int v = 0; v < 8; ++v)
          sA[(mb + v) * LDW + n] = fmaxf(acc[nt][v] + bias, 0.f);
      }
    } else {                                          // jacobian rows: raw v0
      #pragma unroll
      for (int nt = 0; nt < 8; ++nt) {
        const int n = nt * 16 + nb;
        #pragma unroll
        for (int v = 0; v < 8; ++v)
          sA[(mb + v) * LDW + n] = acc[nt][v];
      }
    }
  }
  __syncthreads();
  for (int idx = tid; idx < TB * WIDTH; idx += 256) { // h0 = g0 .* v0 (g0 = y0>0)
    int s = idx >> 7, n = idx & 127;
    float g = sA[s * LDW + n];
    float h = sA[(TB + s) * LDW + n];
    sA[(TB + s) * LDW + n] = (g > 0.f) ? h : 0.f;
  }
  __syncthreads();

  // ============ Phase B: [y0 ; h0] @ W1^T -> a1 | t1raw (128x128) ============
  {
    v8f acc[8]; for (int nt = 0; nt < 8; ++nt) acc[nt] = z8;
    for (int ks = 0; ks < 32; ++ks) {
      v2f a = fragA(sA, LDW, w * 16, ks * 4, lane);
      #pragma unroll
      for (int nt = 0; nt < 8; ++nt) {
        v2f b = fragBT(sW1, LDW, nt * 16, ks * 4, lane);
        acc[nt] = WMMA_F32(a, b, acc[nt]);
      }
    }
    const int mb = w * 16 + 8 * (lane >> 4), nb = lane & 15;
    if (w < 4) {                                      // y1 = relu(a1+b1)
      #pragma unroll
      for (int nt = 0; nt < 8; ++nt) {
        const int n = nt * 16 + nb;
        const float bias = sb1[n];
        #pragma unroll
        for (int v = 0; v < 8; ++v)
          sB[(mb + v) * LDW + n] = fmaxf(acc[nt][v] + bias, 0.f);
      }
    } else {                                          // raw W1 h0
      #pragma unroll
      for (int nt = 0; nt < 8; ++nt) {
        const int n = nt * 16 + nb;
        #pragma unroll
        for (int v = 0; v < 8; ++v)
          sB[(mb + v) * LDW + n] = acc[nt][v];
      }
    }
  }
  __syncthreads();
  for (int idx = tid; idx < TB * WIDTH; idx += 256) { // t1 = g1 .* (W1 h0)
    int s = idx >> 7, n = idx & 127;
    float g = sB[s * LDW + n];
    float t = sB[(TB + s) * LDW + n];
    sB[(TB + s) * LDW + n] = (g > 0.f) ? t : 0.f;
  }
  __syncthreads();

  // ============ Phase C: [y1 ; t1] @ Wcat^T -> head | ldt (128x32) ===========
  {
    v8f acc[2]; acc[0] = z8; acc[1] = z8;
    for (int ks = 0; ks < 32; ++ks) {
      v2f a = fragA(sB, LDW, w * 16, ks * 4, lane);
      #pragma unroll
      for (int nt = 0; nt < 2; ++nt) {
        v2f b = fragBT(sWc, LDW, nt * 16, ks * 4, lane);
        acc[nt] = WMMA_F32(a, b, acc[nt]);
      }
    }
    const int mb = w * 16 + 8 * (lane >> 4), nb = lane & 15;
    if (w < 4) {                                      // fwd heads: raw pre-activation + bias
      #pragma unroll
      for (int nt = 0; nt < 2; ++nt) {
        const int n = nt * 16 + nb;
        const float bias = sbc[n];
        #pragma unroll
        for (int v = 0; v < 8; ++v)
          sH[(mb + v) * 32 + n] = acc[nt][v] + bias;
      }
    } else {                                          // raw ldt (no bias)
      #pragma unroll
      for (int nt = 0; nt < 2; ++nt) {
        const int n = nt * 16 + nb;
        #pragma unroll
        for (int v = 0; v < 8; ++v)
          sH[(mb + v) * 32 + n] = acc[nt][v];
      }
    }
  }
  __syncthreads();

  // ======= Phase D: per-sample 7x7 epilogue (threads 0..63, waves 0-1) =======
  if (tid < TB) {
    const int s = tid;
    float qdv[7];
    for (int k = 0; k < 7; ++k) qdv[k] = sIn[(TB + s) * 8 + k];

    float L[7][7], Ldt[7][7], gd[7];
    for (int i = 0; i < 7; ++i)
      for (int j = 0; j < 7; ++j) { L[i][j] = 0.f; Ldt[i][j] = 0.f; }
    {
      int t = 0;
      for (int i = 0; i < 7; ++i) {
        for (int j = 0; j < i; ++j) {                 // row-major lower-tri fill
          L[i][j]   = sH[s * 32 + 7 + t];
          Ldt[i][j] = sH[(TB + s) * 32 + 7 + t];
          ++t;
        }
        float ad = sH[s * 32 + i];
        float g  = (ad > 0.f) ? 1.f : 0.f;
        gd[i] = g;
        L[i][i]   = fmaxf(ad, 0.f) + EPSV;            // l_diag = relu + eps
        Ldt[i][i] = g * sH[(TB + s) * 32 + i];        // gated diag jacobian
      }
    }
    float u[7], r[7];
    for (int m = 0; m < 7; ++m) {
      float su = 0.f, sr = 0.f;
      for (int i = m; i < 7; ++i) { su += L[i][m] * qdv[i]; sr += Ldt[i][m] * qdv[i]; }
      u[m] = su; r[m] = sr;                           // u = L^T qd, r = Ldt^T qd
    }
    float Tv = 0.f;
    for (int m = 0; m < 7; ++m) Tv += u[m] * u[m];
    Tv *= 0.5f;                                       // T = 0.5 |L^T qd|^2

    float Li[7][7];                                   // Linv by forward substitution
    for (int i = 0; i < 7; ++i) {
      float inv = 1.f / L[i][i];
      for (int j = 0; j < i; ++j) {
        float a = 0.f;
        for (int k = j; k < i; ++k) a += L[i][k] * Li[k][j];
        Li[i][j] = -a * inv;
      }
      Li[i][i] = inv;
    }
    for (int i = 0; i < 7; ++i) {                     // Mdt_qd = L r + Ldt u
      float a = 0.f;
      for (int m = 0; m <= i; ++m) a += L[i][m] * r[m] + Ldt[i][m] * u[m];
      sMd[s * 8 + i] = a;
    }
    // w' in cat order (diag rows gated by gd), padded to 32 -> backward GEMM A
    for (int d = 0; d < 7; ++d) sH[(TB + s) * 32 + d] = gd[d] * qdv[d] * u[d];
    {
      int t = 0;
      for (int i = 0; i < 7; ++i)
        for (int j = 0; j < i; ++j) { sH[(TB + s) * 32 + 7 + t] = qdv[i] * u[j]; ++t; }
    }
    for (int t2 = 28; t2 < 32; ++t2) sH[(TB + s) * 32 + t2] = 0.f;

    if (gb + s < Btot) {
      const size_t gs = (size_t)(gb + s);
      for (int i = 0; i < 7; ++i)
        for (int j = 0; j < 7; ++j) {
          int mn = (i < j) ? i : j, mx = (i > j) ? i : j;
          float mij = 0.f, mi = 0.f;
          for (int m = 0; m <= mn; ++m) mij += L[i][m] * L[j][m];
          for (int k = mx; k < 7; ++k) mi += Li[k][i] * Li[k][j];
          outM [gs * 49 + i * 7 + j] = mij;           // M = L L^T
          outMi[gs * 49 + i * 7 + j] = mi;            // M^-1 = Linv^T Linv
        }
      outT[gs] = Tv;
    }
  }
  __syncthreads();

  // ===== Phase E1: s = w' (64x32) @ Wcat (32x128); t = g1 .* s ==============
  {
    const int mt = w >> 1, ntb = (w & 1) * 4;
    v8f acc[4]; for (int j = 0; j < 4; ++j) acc[j] = z8;
    #pragma unroll
    for (int ks = 0; ks < 8; ++ks) {
      v2f a = fragA(sH + TB * 32, 32, mt * 16, ks * 4, lane);
      #pragma unroll
      for (int j = 0; j < 4; ++j) {
        v2f b = fragBN(sWc, LDW, ks * 4, (ntb + j) * 16, lane);
        acc[j] = WMMA_F32(a, b, acc[j]);
      }
    }
    const int mb = mt * 16 + 8 * (lane >> 4), nb = lane & 15;
    #pragma unroll
    for (int j = 0; j < 4; ++j)
      #pragma unroll
      for (int v = 0; v < 8; ++v) {
        int m = mb + v, n = (ntb + j) * 16 + nb;
        sA[(TB + m) * LDW + n] = (sB[m * LDW + n] > 0.f) ? acc[j][v] : 0.f;
      }
  }
  __syncthreads();

  // ===== Phase E2: z0 = t (64x128) @ W1 (untransposed); z = g0 .* z0 ========
  {
    const int mt = w >> 1, ntb = (w & 1) * 4;
    v8f acc[4]; for (int j = 0; j < 4; ++j) acc[j] = z8;
    for (int ks = 0; ks < 32; ++ks) {
      v2f a = fragA(sA + TB * LDW, LDW, mt * 16, ks * 4, lane);
      #pragma unroll
      for (int j = 0; j < 4; ++j) {
        v2f b = fragBN(sW1, LDW, ks * 4, (ntb + j) * 16, lane);
        acc[j] = WMMA_F32(a, b, acc[j]);
      }
    }
    const int mb = mt * 16 + 8 * (lane >> 4), nb = lane & 15;
    #pragma unroll
    for (int j = 0; j < 4; ++j)
      #pragma unroll
      for (int v = 0; v < 8; ++v) {
        int m = mb + v, n = (ntb + j) * 16 + nb;
        sB[(TB + m) * LDW + n] = (sA[m * LDW + n] > 0.f) ? acc[j][v] : 0.f;
      }
  }
  __syncthreads();

  // ===== Phase E3: 0.5*quad_dq = z (64x128) @ W0; c = Mdt_qd - z@W0 =========
  if (w < 4) {
    v8f acc = z8;
    for (int ks = 0; ks < 32; ++ks) {
      v2f a = fragA(sB + TB * LDW, LDW, w * 16, ks * 4, lane);
      v2f b = fragBN(sW0, 16, ks * 4, 0, lane);
      acc = WMMA_F32(a, b, acc);
    }
    const int mb = w * 16 + 8 * (lane >> 4), nb = lane & 15;
    #pragma unroll
    for (int v = 0; v < 8; ++v) {
      int m = mb + v;
      if (nb < DIMQ && gb + m < Btot)
        outC[(size_t)(gb + m) * DIMQ + nb] = sMd[m * 8 + nb] - acc[v];
    }
  }
}

static const int SMEM_FLOATS =
    128 * LDW + 128 * 16 + 32 * LDW + 128 * 8 + 128 * LDW + 128 * LDW +
    128 * 32 + 64 * 8 + 128 + 128 + 32;  // = 62880 floats = 251,520 bytes

extern "C" void kernel_launch(void* const* d_in, const int* in_sizes, int n_in,
                              void* d_out, int out_size, void* d_ws, size_t ws_size,
                              hipStream_t stream) {
  const float* q   = (const float*)d_in[0];
  const float* qd  = (const float*)d_in[1];
  const float* W0  = (const float*)d_in[2];
  const float* b0  = (const float*)d_in[3];
  const float* W1  = (const float*)d_in[4];
  const float* b1  = (const float*)d_in[5];
  const float* Wlo = (const float*)d_in[6];
  const float* blo = (const float*)d_in[7];
  const float* Wld = (const float*)d_in[8];
  const float* bld = (const float*)d_in[9];

  const int Btot = in_sizes[0] / DIMQ;
  float* outM  = (float*)d_out;
  float* outMi = outM  + (size_t)Btot * 49;
  float* outC  = outMi + (size_t)Btot * 49;
  float* outT  = outC  + (size_t)Btot * DIMQ;

  const int grid = (Btot + TB - 1) / TB;
  delan_fused<<<grid, 256, SMEM_FLOATS * sizeof(float), stream>>>(
      q, qd, W0, b0, W1, b1, Wlo, blo, Wld, bld, outM, outMi, outC, outT, Btot);
}